// GraphCare_85177791414324
// MI455X (gfx1250) — compile-verified
//
#include <hip/hip_runtime.h>
#include <hip/hip_bf16.h>

// ---------------------------------------------------------------------------
// GraphCare forward for MI455X (gfx1250, wave32).
// fp32 end-to-end; dense matmuls via V_WMMA_F32_16X16X4_F32; GCN weight
// matrices staged into LDS with the Tensor Data Mover (TENSOR_LOAD_TO_LDS).
// ---------------------------------------------------------------------------

typedef __attribute__((ext_vector_type(2))) float v2f;
typedef __attribute__((ext_vector_type(8))) float v8f;
typedef __attribute__((ext_vector_type(4))) unsigned int v4u;
typedef __attribute__((ext_vector_type(8))) int v8i;
typedef __attribute__((ext_vector_type(4))) int v4i;

constexpr int Bn = 16, Vv = 20, EC = 24, EP = 16, ED = 24;
constexpr int Dd = 128, OUTD = 512;
constexpr int ET = EC + EP + ED;          // 64 node slots per (b,v)
constexpr int Nn = Bn * Vv * ET;          // 20480 nodes
constexpr int NEe = Nn * 32;              // 655360 edges

#define HAVE_TDM (__has_builtin(__builtin_amdgcn_tensor_load_to_lds) && \
                  __has_builtin(__builtin_amdgcn_s_wait_tensorcnt))

// ---------------------------------------------------------------------------
// Embedding gather: nodes[n][d] from the three tables.
// ---------------------------------------------------------------------------
__global__ void __launch_bounds__(256)
embed_kernel(float* __restrict__ nodes,
             const int* __restrict__ tok_c, const int* __restrict__ tok_p,
             const int* __restrict__ tok_d,
             const float* __restrict__ emb_c, const float* __restrict__ emb_p,
             const float* __restrict__ emb_d) {
  int idx = blockIdx.x * 256 + threadIdx.x;        // N*32 float4 chunks
  if (idx >= Nn * 32) return;
  int c4 = idx & 31;
  int n  = idx >> 5;
  int e  = n & (ET - 1);
  int bv = n >> 6;                                  // b*V + v
  const float* row;
  if (e < EC)           row = emb_c + (size_t)tok_c[bv * EC + e] * Dd;
  else if (e < EC + EP) row = emb_p + (size_t)tok_p[bv * EP + (e - EC)] * Dd;
  else                  row = emb_d + (size_t)tok_d[bv * ED + (e - EC - EP)] * Dd;
  float4 v = ((const float4*)row)[c4];
  ((float4*)(nodes + (size_t)n * Dd))[c4] = v;
}

// ---------------------------------------------------------------------------
// Degree / normalization.
// ---------------------------------------------------------------------------
__global__ void deg_init_kernel(float* __restrict__ deg) {
  int i = blockIdx.x * 256 + threadIdx.x;
  if (i < Nn) deg[i] = 1.0f;                        // self loop
}
__global__ void deg_edge_kernel(float* __restrict__ deg, const int* __restrict__ ei) {
  int e = blockIdx.x * 256 + threadIdx.x;
  if (e < NEe) atomicAdd(&deg[ei[NEe + e]], 1.0f);  // col = edge_index[1]
}
__global__ void dinv_kernel(float* __restrict__ dinv, const float* __restrict__ deg) {
  int i = blockIdx.x * 256 + threadIdx.x;
  if (i < Nn) dinv[i] = rsqrtf(deg[i]);             // deg >= 1 always
}

// ---------------------------------------------------------------------------
// GCN GEMM: Y[Nn,128] = X[Nn,128] @ W[128,128]  (W row-major)
// W staged into LDS via the Tensor Data Mover; B-fragments read from LDS.
// block = 256 (8 waves); each block computes 64 rows (4 M-tiles x 8 N-tiles).
// grid = Nn/64.
// ---------------------------------------------------------------------------
constexpr int GROWS = 64;

__global__ void __launch_bounds__(256)
gcn_gemm_wmma(float* __restrict__ Y, const float* __restrict__ X,
              const float* __restrict__ W) {
  __shared__ float Wlds[Dd * Dd];          // 64 KB, LDS offset 0 (only shared var)

#if HAVE_TDM
  if (threadIdx.x < 32) {
    if (threadIdx.x == 0) {
      // Visible seed store: the real writer of Wlds is the TDM engine, which
      // the optimizer cannot see. Without at least one IR-visible store the
      // compiler folds all Wlds loads to undef and deletes the ds_loads.
      ((volatile float*)Wlds)[0] = W[0];
    }
    // ---- Tensor DMA descriptor (D#): 2-D, 128x128, 4-byte elements ----
    unsigned long long ga = (unsigned long long)(size_t)W;
    v4u g0;
    g0.x = 1u;                                          // count=1, user mode
    g0.y = 0u;                                          // lds_addr = 0 (Wlds)
    g0.z = (unsigned)(ga & 0xFFFFFFFFu);                // global_addr[31:0]
    g0.w = (unsigned)((ga >> 32) & 0x01FFFFFFu)         // global_addr[56:32]
         | (2u << 30);                                  // type = 2 ("image")
    v8i g1;
    g1[0] = 0x00020000;          // workgroup_mask=0, data_size=2 (4B)
    g1[1] = (int)(128u << 16);   // tensor_dim0 = 128 (elements)
    g1[2] = (int)(128u << 16);   // tensor_dim1 = 128
    g1[3] = (int)(128u << 16);   // tile_dim0   = 128
    g1[4] = 128;                 // tile_dim1   = 128, tile_dim2 = 0
    g1[5] = 128;                 // tensor_dim0_stride = 128 (elements)
    g1[6] = 0;
    g1[7] = 0;
    v4i g2 = {0, 0, 0, 0};       // unused (2-D tensor)
    v4i g3 = {0, 0, 0, 0};
#if __clang_major__ >= 23
    v8i g4 = {0, 0, 0, 0, 0, 0, 0, 0};
    __builtin_amdgcn_tensor_load_to_lds(g0, g1, g2, g3, g4, 0);
#else
    __builtin_amdgcn_tensor_load_to_lds(g0, g1, g2, g3, 0);
#endif
    __builtin_amdgcn_s_wait_tensorcnt(0);               // s_wait_tensorcnt 0
  }
#else
  for (int i = threadIdx.x; i < Dd * Dd / 4; i += 256)
    ((float4*)Wlds)[i] = ((const float4*)W)[i];
#endif
  // All threads: LDS was mutated behind the compiler's back (TDM DMA) —
  // force it to re-load Wlds after the barrier.
  asm volatile("" ::: "memory");
  __syncthreads();

  const int wave = threadIdx.x >> 5;
  const int lane = threadIdx.x & 31;
  const int mr   = lane & 15;
  const int koff = (lane >> 4) << 1;                    // 0 or 2
  const int mh   = (lane >> 4) << 3;                    // 0 or 8
  const int n0   = wave << 4;
  const int mbase = blockIdx.x * GROWS;

#pragma unroll
  for (int mt = 0; mt < GROWS / 16; ++mt) {
    const int m0 = mbase + mt * 16;
    const float* xrow = X + (size_t)(m0 + mr) * Dd;
    v8f c = {};
    for (int k = 0; k < Dd; k += 4) {
      v2f a, b;
      a.x = xrow[k + koff];
      a.y = xrow[k + koff + 1];
      b.x = Wlds[(k + koff) * Dd + n0 + mr];
      b.y = Wlds[(k + koff + 1) * Dd + n0 + mr];
      c = __builtin_amdgcn_wmma_f32_16x16x4_f32(false, a, false, b, (short)0, c,
                                                false, false);
    }
#pragma unroll
    for (int r = 0; r < 8; ++r)
      Y[(size_t)(m0 + mh + r) * Dd + n0 + mr] = c[r];
  }
}

// ---------------------------------------------------------------------------
// GEMM  Y[M,N] = X[M,K] @ Wt[N,K]^T + bias[N]   (Wt row-major N x K).
// grid = (M/16, N/128)
// ---------------------------------------------------------------------------
__global__ void __launch_bounds__(256)
gemm_nt_bias_wmma(float* __restrict__ Y, const float* __restrict__ X,
                  const float* __restrict__ Wt, const float* __restrict__ bias,
                  int M, int N, int K) {
  const int wave = threadIdx.x >> 5;
  const int lane = threadIdx.x & 31;
  const int m0 = blockIdx.x * 16;
  const int n0 = (blockIdx.y * 8 + wave) * 16;
  if (m0 >= M || n0 >= N) return;
  const int mr   = lane & 15;
  const int koff = (lane >> 4) << 1;
  const float* xrow = X + (size_t)(m0 + mr) * K;
  const float* wrow = Wt + (size_t)(n0 + mr) * K;
  v8f c = {};
  for (int k = 0; k < K; k += 4) {
    v2f a, b;
    a.x = xrow[k + koff];
    a.y = xrow[k + koff + 1];
    b.x = wrow[k + koff];
    b.y = wrow[k + koff + 1];
    c = __builtin_amdgcn_wmma_f32_16x16x4_f32(false, a, false, b, (short)0, c,
                                              false, false);
  }
  const int mh = (lane >> 4) << 3;
  const float bb = bias[n0 + mr];
#pragma unroll
  for (int r = 0; r < 8; ++r)
    Y[(size_t)(m0 + mh + r) * N + n0 + mr] = c[r] + bb;
}

// ---------------------------------------------------------------------------
// FC: Y[M,N] = relu(X[M,K]) @ W[K,N] + bias[N]
// ---------------------------------------------------------------------------
__global__ void __launch_bounds__(256)
fc_wmma(float* __restrict__ Y, const float* __restrict__ X,
        const float* __restrict__ W, const float* __restrict__ bias,
        int M, int N, int K) {
  const int wave = threadIdx.x >> 5;
  const int lane = threadIdx.x & 31;
  const int m0 = blockIdx.x * 16;
  const int n0 = (blockIdx.y * 8 + wave) * 16;
  if (m0 >= M || n0 >= N) return;
  const int mr   = lane & 15;
  const int koff = (lane >> 4) << 1;
  const float* xrow = X + (size_t)(m0 + mr) * K;
  v8f c = {};
  for (int k = 0; k < K; k += 4) {
    v2f a, b;
    a.x = fmaxf(xrow[k + koff], 0.0f);
    a.y = fmaxf(xrow[k + koff + 1], 0.0f);
    const float* wp = W + (size_t)(k + koff) * N + n0 + mr;
    b.x = wp[0];
    b.y = wp[N];
    c = __builtin_amdgcn_wmma_f32_16x16x4_f32(false, a, false, b, (short)0, c,
                                              false, false);
  }
  const int mh = (lane >> 4) << 3;
  const float bb = bias[n0 + mr];
#pragma unroll
  for (int r = 0; r < 8; ++r)
    Y[(size_t)(m0 + mh + r) * N + n0 + mr] = c[r] + bb;
}

// ---------------------------------------------------------------------------
// Aggregation: out[n] = sum over edges (col==n) of h[row]*dinv[row]*dinv[col]
// Self-loop handled by init; edges via f32 atomics.
// ---------------------------------------------------------------------------
__global__ void __launch_bounds__(256)
agg_init_kernel(float* __restrict__ out, const float* __restrict__ h,
                const float* __restrict__ dinv) {
  int idx = blockIdx.x * 256 + threadIdx.x;        // N*32 chunks
  if (idx >= Nn * 32) return;
  int c = idx & 31;
  int n = idx >> 5;
  float w = dinv[n]; w *= w;
  float4 v = ((const float4*)(h + (size_t)n * Dd))[c];
  v.x *= w; v.y *= w; v.z *= w; v.w *= w;
  ((float4*)(out + (size_t)n * Dd))[c] = v;
}

__global__ void __launch_bounds__(256)
agg_edge_kernel(float* __restrict__ out, const float* __restrict__ h,
                const float* __restrict__ dinv, const int* __restrict__ ei) {
  int idx = blockIdx.x * 256 + threadIdx.x;        // NE*32 chunks
  if (idx >= NEe * 32) return;
  int c = idx & 31;
  int e = idx >> 5;
  int row = ei[e];
  int col = ei[NEe + e];
  float norm = dinv[row] * dinv[col];
  float4 v = ((const float4*)(h + (size_t)row * Dd))[c];
  float* o = out + (size_t)col * Dd + c * 4;
  atomicAdd(o + 0, v.x * norm);
  atomicAdd(o + 1, v.y * norm);
  atomicAdd(o + 2, v.z * norm);
  atomicAdd(o + 3, v.w * norm);
}

__global__ void __launch_bounds__(256)
bias_act_kernel(float* __restrict__ dst, const float* __restrict__ src,
                const float* __restrict__ bias, int relu) {
  int idx = blockIdx.x * 256 + threadIdx.x;        // N*128
  if (idx >= Nn * Dd) return;
  float v = src[idx] + bias[idx & (Dd - 1)];
  if (relu) v = fmaxf(v, 0.0f);
  dst[idx] = v;
}

// ---------------------------------------------------------------------------
// Masked visit sum: seq_X[b][e][d] = sum_v h[node(b,v,e)][d] * (tok != 0)
// ---------------------------------------------------------------------------
__global__ void __launch_bounds__(256)
visit_sum_kernel(float* __restrict__ sc, float* __restrict__ sp,
                 float* __restrict__ sd, const float* __restrict__ h,
                 const int* __restrict__ tc, const int* __restrict__ tp,
                 const int* __restrict__ td) {
  int tid = blockIdx.x * 256 + threadIdx.x;        // B*64*128
  if (tid >= Bn * ET * Dd) return;
  int d = tid & (Dd - 1);
  int e = (tid >> 7) & (ET - 1);
  int b = tid >> 13;
  float acc = 0.0f;
  if (e < EC) {
    for (int v = 0; v < Vv; ++v)
      if (tc[(b * Vv + v) * EC + e] != 0)
        acc += h[(size_t)((b * Vv + v) * ET + e) * Dd + d];
    sc[((b * EC + e) * Dd) + d] = acc;
  } else if (e < EC + EP) {
    int ep = e - EC;
    for (int v = 0; v < Vv; ++v)
      if (tp[(b * Vv + v) * EP + ep] != 0)
        acc += h[(size_t)((b * Vv + v) * ET + e) * Dd + d];
    sp[((b * EP + ep) * Dd) + d] = acc;
  } else {
    int ed = e - EC - EP;
    for (int v = 0; v < Vv; ++v)
      if (td[(b * Vv + v) * ED + ed] != 0)
        acc += h[(size_t)((b * Vv + v) * ET + e) * Dd + d];
    sd[((b * ED + ed) * Dd) + d] = acc;
  }
}

// ---------------------------------------------------------------------------
// GRU recurrence. gi (= x@Wih^T + bih) precomputed; per step compute
// gh = h@Whh^T + bhh via WMMA (h in LDS), then elementwise gates.
// One block per GRU stream (blockIdx.x = 0..2), 256 threads = 8 waves.
// ---------------------------------------------------------------------------
struct GruArgs {
  const float* gi[3];
  const float* whh[3];
  const float* bhh[3];
  int T[3];
  float* pe;                 // [16][384]
};

__global__ void __launch_bounds__(256)
gru_kernel(GruArgs args) {
  __shared__ float hsh[Bn * Dd];          // 8 KB
  __shared__ float ghsh[Bn * 3 * Dd];     // 24 KB
  const int g = blockIdx.x;
  const float* gi  = args.gi[g];
  const float* whh = args.whh[g];
  const float* bhh = args.bhh[g];
  const int T = args.T[g];

  for (int i = threadIdx.x; i < Bn * Dd; i += 256) hsh[i] = 0.0f;
  __syncthreads();

  const int wave = threadIdx.x >> 5;
  const int lane = threadIdx.x & 31;
  const int mr   = lane & 15;
  const int koff = (lane >> 4) << 1;
  const int mh   = (lane >> 4) << 3;

  for (int t = 0; t < T; ++t) {
    // gh = h @ whh^T + bhh  :  M=16, N=384 (24 tiles over 8 waves), K=128
    for (int nt = wave; nt < 24; nt += 8) {
      const int n0 = nt << 4;
      const float* wrow = whh + (size_t)(n0 + mr) * Dd;
      v8f c = {};
      for (int k = 0; k < Dd; k += 4) {
        v2f a, b;
        a.x = hsh[mr * Dd + k + koff];
        a.y = hsh[mr * Dd + k + koff + 1];
        b.x = wrow[k + koff];
        b.y = wrow[k + koff + 1];
        c = __builtin_amdgcn_wmma_f32_16x16x4_f32(false, a, false, b, (short)0,
                                                  c, false, false);
      }
      const float bb = bhh[n0 + mr];
#pragma unroll
      for (int r = 0; r < 8; ++r)
        ghsh[(mh + r) * 384 + n0 + mr] = c[r] + bb;
    }
    __syncthreads();
    // gates
    for (int i = threadIdx.x; i < Bn * Dd; i += 256) {
      int b = i >> 7, d = i & (Dd - 1);
      const float* girow = gi + ((size_t)b * T + t) * 384;
      float ir = girow[d], iz = girow[Dd + d], in = girow[2 * Dd + d];
      float hr = ghsh[b * 384 + d];
      float hz = ghsh[b * 384 + Dd + d];
      float hn = ghsh[b * 384 + 2 * Dd + d];
      float r = 1.0f / (1.0f + __expf(-(ir + hr)));
      float z = 1.0f / (1.0f + __expf(-(iz + hz)));
      float n = tanhf(in + r * hn);
      hsh[i] = (1.0f - z) * n + z * hsh[i];
    }
    __syncthreads();
  }
  for (int i = threadIdx.x; i < Bn * Dd; i += 256) {
    int b = i >> 7, d = i & (Dd - 1);
    args.pe[b * 384 + g * Dd + d] = hsh[i];
  }
}

// ---------------------------------------------------------------------------
// Host launcher
// ---------------------------------------------------------------------------
extern "C" void kernel_launch(void* const* d_in, const int* in_sizes, int n_in,
                              void* d_out, int out_size, void* d_ws, size_t ws_size,
                              hipStream_t stream) {
  const int*   tok_c = (const int*)d_in[0];
  const int*   tok_p = (const int*)d_in[1];
  const int*   tok_d = (const int*)d_in[2];
  const int*   ei    = (const int*)d_in[3];
  const float* emb_c = (const float*)d_in[4];
  const float* emb_p = (const float*)d_in[5];
  const float* emb_d = (const float*)d_in[6];
  const float* g1w = (const float*)d_in[7];
  const float* g1b = (const float*)d_in[8];
  const float* g2w = (const float*)d_in[9];
  const float* g2b = (const float*)d_in[10];
  const float* wih_c = (const float*)d_in[11];
  const float* whh_c = (const float*)d_in[12];
  const float* bih_c = (const float*)d_in[13];
  const float* bhh_c = (const float*)d_in[14];
  const float* wih_p = (const float*)d_in[15];
  const float* whh_p = (const float*)d_in[16];
  const float* bih_p = (const float*)d_in[17];
  const float* bhh_p = (const float*)d_in[18];
  const float* wih_d = (const float*)d_in[19];
  const float* whh_d = (const float*)d_in[20];
  const float* bih_d = (const float*)d_in[21];
  const float* bhh_d = (const float*)d_in[22];
  const float* fcw = (const float*)d_in[23];
  const float* fcb = (const float*)d_in[24];
  float* out = (float*)d_out;

  float* ws = (float*)d_ws;
  size_t off = 0;
  auto alloc = [&](size_t nfl) { float* p = ws + off; off += nfl; return p; };
  float* nodes = alloc((size_t)Nn * Dd);
  float* tmpA  = alloc((size_t)Nn * Dd);
  float* tmpB  = alloc((size_t)Nn * Dd);
  float* deg   = alloc(Nn);
  float* dinv  = alloc(Nn);
  float* seq_c = alloc((size_t)Bn * EC * Dd);
  float* seq_p = alloc((size_t)Bn * EP * Dd);
  float* seq_d = alloc((size_t)Bn * ED * Dd);
  float* gi_c  = alloc((size_t)Bn * EC * 384);
  float* gi_p  = alloc((size_t)Bn * EP * 384);
  float* gi_d  = alloc((size_t)Bn * ED * 384);
  float* pe    = alloc((size_t)Bn * 384);

  const int TB = 256;
  // 1. embeddings
  embed_kernel<<<(Nn * 32 + TB - 1) / TB, TB, 0, stream>>>(
      nodes, tok_c, tok_p, tok_d, emb_c, emb_p, emb_d);
  // 2. degree / norm (shared by both layers)
  deg_init_kernel<<<(Nn + TB - 1) / TB, TB, 0, stream>>>(deg);
  deg_edge_kernel<<<(NEe + TB - 1) / TB, TB, 0, stream>>>(deg, ei);
  dinv_kernel<<<(Nn + TB - 1) / TB, TB, 0, stream>>>(dinv, deg);
  // 3. GCN layer 1
  gcn_gemm_wmma<<<Nn / GROWS, TB, 0, stream>>>(tmpA, nodes, g1w);
  agg_init_kernel<<<(Nn * 32 + TB - 1) / TB, TB, 0, stream>>>(tmpB, tmpA, dinv);
  agg_edge_kernel<<<(NEe * 32 + TB - 1) / TB, TB, 0, stream>>>(tmpB, tmpA, dinv, ei);
  bias_act_kernel<<<(Nn * Dd + TB - 1) / TB, TB, 0, stream>>>(nodes, tmpB, g1b, 1);
  // 4. GCN layer 2
  gcn_gemm_wmma<<<Nn / GROWS, TB, 0, stream>>>(tmpA, nodes, g2w);
  agg_init_kernel<<<(Nn * 32 + TB - 1) / TB, TB, 0, stream>>>(tmpB, tmpA, dinv);
  agg_edge_kernel<<<(NEe * 32 + TB - 1) / TB, TB, 0, stream>>>(tmpB, tmpA, dinv, ei);
  bias_act_kernel<<<(Nn * Dd + TB - 1) / TB, TB, 0, stream>>>(nodes, tmpB, g2b, 0);
  // 5. masked visit sums
  visit_sum_kernel<<<(Bn * ET * Dd + TB - 1) / TB, TB, 0, stream>>>(
      seq_c, seq_p, seq_d, nodes, tok_c, tok_p, tok_d);
  // 6. GRU input-gate GEMMs (parallel over all timesteps)
  gemm_nt_bias_wmma<<<dim3(Bn * EC / 16, 3), TB, 0, stream>>>(gi_c, seq_c, wih_c, bih_c, Bn * EC, 384, Dd);
  gemm_nt_bias_wmma<<<dim3(Bn * EP / 16, 3), TB, 0, stream>>>(gi_p, seq_p, wih_p, bih_p, Bn * EP, 384, Dd);
  gemm_nt_bias_wmma<<<dim3(Bn * ED / 16, 3), TB, 0, stream>>>(gi_d, seq_d, wih_d, bih_d, Bn * ED, 384, Dd);
  // 7. GRU recurrences (3 streams in parallel blocks)
  GruArgs ga;
  ga.gi[0] = gi_c;  ga.gi[1] = gi_p;  ga.gi[2] = gi_d;
  ga.whh[0] = whh_c; ga.whh[1] = whh_p; ga.whh[2] = whh_d;
  ga.bhh[0] = bhh_c; ga.bhh[1] = bhh_p; ga.bhh[2] = bhh_d;
  ga.T[0] = EC; ga.T[1] = EP; ga.T[2] = ED;
  ga.pe = pe;
  gru_kernel<<<3, TB, 0, stream>>>(ga);
  // 8. final FC: logits = relu(pe) @ fc_w + fc_b
  fc_wmma<<<dim3(Bn / 16, OUTD / 128), TB, 0, stream>>>(out, pe, fcw, fcb, Bn, OUTD, 3 * Dd);
}